// LocalMamba2DModel_33457795235973
// MI455X (gfx1250) — compile-verified
//
#include <hip/hip_runtime.h>
#include <hip/hip_bf16.h>

// ---------------------------------------------------------------------------
// LocalMamba2D forward for gfx1250 (MI455X).
//  - big GEMMs: workgroup-tiled WMMA (f32 16x16x4) with TDM (tensor_load_to_lds)
//    double-buffered LDS staging, s_wait_tensorcnt + barriers
//  - small/gathered GEMMs: one-wave-per-tile WMMA
//  - selective scan: fused 4-direction register-resident recurrence
// ---------------------------------------------------------------------------

typedef __attribute__((ext_vector_type(2))) float v2f;
typedef __attribute__((ext_vector_type(8))) float v8f;

#define CB  8        // batch
#define CCIN 3
#define CIMG 128
#define CDM 512
#define CDI 1024
#define CN  16
#define CR  32
#define CK  4
#define CMW 256
#define CMFF 768
#define CL  1024     // 32*32 tokens per image
#define CROWS (CB*CL)   // 8192 token rows

// macro-tile geometry for the TDM GEMM
#define MB 128       // rows per workgroup tile
#define NB 64        // cols per workgroup tile
#define KC 32        // K panel depth
#define LSTR 34      // padded LDS row stride in dwords (32 data + 2 pad)

// ---------------------------------------------------------------------------
// TDM descriptor issue: 2D tile (rows x cols fp32) from global into LDS,
// row padding 2 dwords per 32 dwords (-> LDS stride 34 dwords).
// Packed per CDNA5 ISA 8.3/8.4.  This toolchain exposes the 6-arg builtin:
//   (uint32x4 g0, int32x8 g1, int32x4 g2, int32x4 g3, int32x8 zero, i32 cpol)
// ---------------------------------------------------------------------------
#if __has_builtin(__builtin_amdgcn_tensor_load_to_lds)
#define USE_TDM 1
typedef unsigned int v4ui __attribute__((ext_vector_type(4)));
typedef int v8i __attribute__((ext_vector_type(8)));
typedef int v4i __attribute__((ext_vector_type(4)));

__device__ __forceinline__ void tdm_load_2d(const float* gptr, unsigned lds_byte,
                                            int rows, int cols,
                                            unsigned long long row_stride_elems,
                                            unsigned tdim0, unsigned tdim1)
{
    unsigned long long ga = (unsigned long long)(uintptr_t)gptr;
    v4ui g0;
    g0[0] = 1u;                                            // count=1, user desc
    g0[1] = lds_byte;                                      // lds_addr (bytes)
    g0[2] = (unsigned)(ga & 0xFFFFFFFFull);                // global_addr[31:0]
    g0[3] = (unsigned)((ga >> 32) & 0x1FFFFFFull)          // global_addr[56:32]
          | (2u << 30);                                    // type=2 (image)
    v8i g1;
    // data_size=2 (4B), pad_enable, pad_interval=4 (32 dw), pad_amount=1 (2 dw)
    g1[0] = (int)((2u << 16) | (1u << 20) | (4u << 22) | (1u << 25));
    g1[1] = (int)((tdim0 & 0xFFFFu) << 16);                          // tensor_dim0 lo16
    g1[2] = (int)(((tdim0 >> 16) & 0xFFFFu) | ((tdim1 & 0xFFFFu) << 16));
    g1[3] = (int)(((tdim1 >> 16) & 0xFFFFu) | (((unsigned)cols & 0xFFFFu) << 16)); // tile_dim0
    g1[4] = (int)((unsigned)rows & 0xFFFFu);                         // tile_dim1 (tile_dim2=0)
    g1[5] = (int)(row_stride_elems & 0xFFFFFFFFull);                 // dim0_stride lo32
    g1[6] = (int)((row_stride_elems >> 32) & 0xFFFFull);             // dim0_stride hi16
    g1[7] = 0;
    v4i z4 = {0, 0, 0, 0};
    v8i z8 = {0, 0, 0, 0, 0, 0, 0, 0};
    __builtin_amdgcn_tensor_load_to_lds(g0, g1, z4, z4, z8, 0);
}
#else
#define USE_TDM 0
#endif

// ---------------------------------------------------------------------------
// Workgroup-tiled WMMA GEMM: C[M,N] (+)= A[M,K] * W[N,K]^T  (row-major fp32)
// block = 256 threads (8 waves); each wave computes a 16x64 strip of the
// 128x64 macro-tile.  A/B panels staged in LDS by TDM (double-buffered).
// M % 128 == 0, N % 64 == 0, K % 32 == 0.
// ---------------------------------------------------------------------------
__global__ __launch_bounds__(256)
void wmma_gemm_tdm_kernel(const float* __restrict__ A,
                          const float* __restrict__ W,
                          float* __restrict__ C,
                          int M, int N, int K, int acc)
{
    __shared__ float As[2][MB * LSTR];
    __shared__ float Bs[2][NB * LSTR];

    int tid  = threadIdx.x;
    int wave = tid >> 5;
    int lane = tid & 31;
    int hf   = lane >> 4;      // selects K pair / M+8 half
    int mn   = lane & 15;
    int n0   = blockIdx.x * NB;
    int m0   = blockIdx.y * MB;
    int mw   = wave * 16;      // this wave's row offset inside macro tile

    v8f c[4] = {};
    if (acc) {
#pragma unroll
        for (int nt = 0; nt < 4; ++nt)
#pragma unroll
            for (int r = 0; r < 8; ++r)
                c[nt][r] = C[(size_t)(m0 + mw + r + 8*hf) * N + n0 + nt*16 + mn];
    }

    int npanel = K / KC;

#if USE_TDM
    unsigned ldsA0 = (unsigned)(uintptr_t)&As[0][0];
    unsigned ldsA1 = (unsigned)(uintptr_t)&As[1][0];
    unsigned ldsB0 = (unsigned)(uintptr_t)&Bs[0][0];
    unsigned ldsB1 = (unsigned)(uintptr_t)&Bs[1][0];
    if (wave == 0) {
        tdm_load_2d(A + (size_t)m0 * K, ldsA0, MB, KC, (unsigned long long)K,
                    (unsigned)K, (unsigned)M);
        tdm_load_2d(W + (size_t)n0 * K, ldsB0, NB, KC, (unsigned long long)K,
                    (unsigned)K, (unsigned)N);
    }
#else
    // cooperative fallback staging of panel 0
    for (int idx = tid; idx < MB * KC; idx += 256) {
        int r = idx / KC, cc = idx % KC;
        As[0][r * LSTR + cc] = A[(size_t)(m0 + r) * K + cc];
    }
    for (int idx = tid; idx < NB * KC; idx += 256) {
        int r = idx / KC, cc = idx % KC;
        Bs[0][r * LSTR + cc] = W[(size_t)(n0 + r) * K + cc];
    }
#endif

    for (int p = 0; p < npanel; ++p) {
        int buf = p & 1;
#if USE_TDM
        if (wave == 0) __builtin_amdgcn_s_wait_tensorcnt(0);
#endif
        __syncthreads();
        if (p + 1 < npanel) {
            int kc = (p + 1) * KC;
#if USE_TDM
            if (wave == 0) {
                unsigned la = (buf == 0) ? ldsA1 : ldsA0;
                unsigned lb = (buf == 0) ? ldsB1 : ldsB0;
                tdm_load_2d(A + (size_t)m0 * K + kc, la, MB, KC,
                            (unsigned long long)K, (unsigned)K, (unsigned)M);
                tdm_load_2d(W + (size_t)n0 * K + kc, lb, NB, KC,
                            (unsigned long long)K, (unsigned)K, (unsigned)N);
            }
#else
            int nb = buf ^ 1;
            for (int idx = tid; idx < MB * KC; idx += 256) {
                int r = idx / KC, cc = idx % KC;
                As[nb][r * LSTR + cc] = A[(size_t)(m0 + r) * K + kc + cc];
            }
            for (int idx = tid; idx < NB * KC; idx += 256) {
                int r = idx / KC, cc = idx % KC;
                Bs[nb][r * LSTR + cc] = W[(size_t)(n0 + r) * K + kc + cc];
            }
#endif
        }
        const float* Ab = As[buf];
        const float* Bb = Bs[buf];
#pragma unroll
        for (int kk = 0; kk < KC; kk += 4) {
            int kb = kk + hf * 2;
            v2f a;
            a.x = Ab[(mw + mn) * LSTR + kb];
            a.y = Ab[(mw + mn) * LSTR + kb + 1];
#pragma unroll
            for (int nt = 0; nt < 4; ++nt) {
                v2f b;
                b.x = Bb[(nt*16 + mn) * LSTR + kb];
                b.y = Bb[(nt*16 + mn) * LSTR + kb + 1];
                c[nt] = __builtin_amdgcn_wmma_f32_16x16x4_f32(false, a, false, b,
                                                              (short)0, c[nt],
                                                              false, false);
            }
        }
        __syncthreads();
    }

#pragma unroll
    for (int nt = 0; nt < 4; ++nt)
#pragma unroll
        for (int r = 0; r < 8; ++r)
            C[(size_t)(m0 + mw + r + 8*hf) * N + n0 + nt*16 + mn] = c[nt][r];
}

// ---------------------------------------------------------------------------
// Simple WMMA GEMM (one wave per 16x16 tile) for small / odd-shaped GEMMs.
// ---------------------------------------------------------------------------
__global__ void wmma_gemm_kernel(const float* __restrict__ A,
                                 const float* __restrict__ W,
                                 float* __restrict__ C,
                                 int M, int N, int K, int acc)
{
    int lane = threadIdx.x;
    int hf   = lane >> 4;
    int mn   = lane & 15;
    int n0   = blockIdx.x * 16;
    int m0   = blockIdx.y * 16;

    v8f c = {};
    if (acc) {
#pragma unroll
        for (int r = 0; r < 8; ++r)
            c[r] = C[(size_t)(m0 + r + 8*hf) * N + n0 + mn];
    }

    const float* Arow = A + (size_t)(m0 + mn) * K;
    const float* Wrow = W + (size_t)(n0 + mn) * K;

    for (int k = 0; k < K; k += 4) {
        int kb = k + hf * 2;
        __builtin_prefetch(Arow + kb + 32, 0, 0);
        v2f a, b;
        a.x = Arow[kb]; a.y = Arow[kb + 1];
        b.x = Wrow[kb]; b.y = Wrow[kb + 1];
        c = __builtin_amdgcn_wmma_f32_16x16x4_f32(false, a, false, b,
                                                  (short)0, c, false, false);
    }

#pragma unroll
    for (int r = 0; r < 8; ++r)
        C[(size_t)(m0 + r + 8*hf) * N + n0 + mn] = c[r];
}

// ---------------------------------------------------------------------------
// x_dbl WMMA GEMM with per-direction row gather:
//   xdbl[b,k,l,c] = sum_d xconv[b, h(l,k), w(l,k), d] * x_proj_w[k, c, d]
// ---------------------------------------------------------------------------
__global__ void xdbl_wmma_kernel(const float* __restrict__ xconv,
                                 const float* __restrict__ xprojw,
                                 float* __restrict__ xdbl)
{
    int lane = threadIdx.x;
    int hf   = lane >> 4;
    int mn   = lane & 15;
    int n0   = blockIdx.x * 16;
    int tm   = blockIdx.y;
    int kd   = blockIdx.z;

    int rm = tm * 16 + mn;
    int bb = rm >> 10;
    int l  = rm & 1023;
    int ld = (kd & 2) ? (1023 - l) : l;
    int hh = (kd & 1) ? (ld & 31) : (ld >> 5);
    int ww = (kd & 1) ? (ld >> 5) : (ld & 31);
    const float* Arow = xconv + ((size_t)((bb*32 + hh)*32 + ww) << 10);
    const float* Wrow = xprojw + (size_t)(kd * 64 + n0 + mn) * CDI;

    v8f c = {};
    for (int k = 0; k < CDI; k += 4) {
        int kb = k + hf * 2;
        v2f a, b;
        a.x = Arow[kb]; a.y = Arow[kb + 1];
        b.x = Wrow[kb]; b.y = Wrow[kb + 1];
        c = __builtin_amdgcn_wmma_f32_16x16x4_f32(false, a, false, b,
                                                  (short)0, c, false, false);
    }

    int bb0   = (tm * 16) >> 10;
    int lbase = (tm * 16) & 1023;
#pragma unroll
    for (int r = 0; r < 8; ++r) {
        int lr = lbase + r + 8*hf;
        xdbl[((size_t)(bb0*4 + kd) * 1024 + lr) * 64 + n0 + mn] = c[r];
    }
}

// ---------------------------------------------------------------------------
__global__ void time_embed_kernel(const float* __restrict__ sigma,
                                  const float* __restrict__ tfw,
                                  float* __restrict__ te)
{
    int idx = blockIdx.x * blockDim.x + threadIdx.x;
    if (idx >= CB * 128) return;
    int i = idx >> 7, j = idx & 127;
    float c = __logf(sigma[i]) * 0.25f;
    float f = 6.283185307179586f * c * tfw[j];
    te[i * 256 + j]       = __cosf(f);
    te[i * 256 + 128 + j] = __sinf(f);
}

__global__ void small_gemm_kernel(const float* __restrict__ A,
                                  const float* __restrict__ W,
                                  const float* __restrict__ skip,
                                  float* __restrict__ C,
                                  int M, int N, int K, float addc)
{
    int idx = blockIdx.x * blockDim.x + threadIdx.x;
    if (idx >= M * N) return;
    int m = idx / N, n = idx % N;
    const float* a = A + (size_t)m * K;
    const float* w = W + (size_t)n * K;
    float acc = addc;
    for (int k = 0; k < K; ++k) acc += a[k] * w[k];
    if (skip) acc += skip[idx];
    C[idx] = acc;
}

__global__ void rmsnorm_kernel(const float* __restrict__ x,
                               const float* __restrict__ g,
                               float* __restrict__ y,
                               int width, int rows_per_b)
{
    int row = blockIdx.x;
    int tid = threadIdx.x;
    __shared__ float red[256];
    const float* xr = x + (size_t)row * width;
    float s = 0.f;
    for (int d = tid; d < width; d += 256) { float t = xr[d]; s += t * t; }
    red[tid] = s; __syncthreads();
    for (int o = 128; o > 0; o >>= 1) {
        if (tid < o) red[tid] += red[tid + o];
        __syncthreads();
    }
    float inv = rsqrtf(red[0] / (float)width + 1e-6f);
    const float* gp = g + (rows_per_b ? (size_t)(row / rows_per_b) * width : 0);
    float* yr = y + (size_t)row * width;
    for (int d = tid; d < width; d += 256) yr[d] = xr[d] * gp[d] * inv;
}

__global__ void glu_gelu_kernel(const float* __restrict__ u,
                                float* __restrict__ h, int M, int F)
{
    int idx = blockIdx.x * blockDim.x + threadIdx.x;
    if (idx >= M * F) return;
    int m = idx / F, j = idx % F;
    float a = u[(size_t)m * 2 * F + j];
    float g = u[(size_t)m * 2 * F + F + j];
    float t = 0.5f * g * (1.f + tanhf(0.7978845608028654f *
                                      (g + 0.044715f * g * g * g)));
    h[idx] = a * t;
}

__global__ void patch_gather_kernel(const float* __restrict__ x,
                                    float* __restrict__ Am)
{
    int idx = blockIdx.x * blockDim.x + threadIdx.x;
    if (idx >= CROWS * 48) return;
    int f = idx % 48, row = idx / 48;
    int b = row >> 10, h = (row >> 5) & 31, w = row & 31;
    int pi = f / 12, pj = (f / 3) % 4, ci = f % 3;
    Am[idx] = x[((size_t)(b * CCIN + ci) * CIMG + (h * 4 + pi)) * CIMG + (w * 4 + pj)];
}

__global__ void patch_scatter_kernel(const float* __restrict__ ysplit,
                                     float* __restrict__ out)
{
    int idx = blockIdx.x * blockDim.x + threadIdx.x;
    if (idx >= CROWS * 48) return;
    int f = idx % 48, row = idx / 48;
    int b = row >> 10, h = (row >> 5) & 31, w = row & 31;
    int pi = f / 12, pj = (f / 3) % 4, ci = f % 3;
    out[((size_t)(b * CCIN + ci) * CIMG + (h * 4 + pi)) * CIMG + (w * 4 + pj)] = ysplit[idx];
}

__global__ void conv_silu_kernel(const float* __restrict__ xz,
                                 const float* __restrict__ cw,
                                 const float* __restrict__ cb,
                                 float* __restrict__ xconv)
{
    int idx = blockIdx.x * blockDim.x + threadIdx.x;
    if (idx >= CB * CL * CDI) return;
    int d = idx & 1023;
    int w = (idx >> 10) & 31;
    int h = (idx >> 15) & 31;
    int b = idx >> 20;
    float acc = cb[d];
#pragma unroll
    for (int ki = 0; ki < 3; ++ki) {
        int hy = h + ki - 1;
        if (hy < 0 || hy > 31) continue;
#pragma unroll
        for (int kj = 0; kj < 3; ++kj) {
            int wx = w + kj - 1;
            if (wx < 0 || wx > 31) continue;
            acc += xz[((size_t)(b * 1024 + hy * 32 + wx)) * 2048 + d]
                 * cw[d * 9 + ki * 3 + kj];
        }
    }
    xconv[idx] = acc / (1.f + __expf(-acc));  // silu
}

// ---------------------------------------------------------------------------
// Selective scan, 4 directions fused (deterministic: each thread owns one
// (b,d) and loops the directions, so cross-merge accumulation is race-free).
// ---------------------------------------------------------------------------
__global__ void scan_kernel(const float* __restrict__ xconv,
                            const float* __restrict__ xdbl,
                            const float* __restrict__ dtw_all,
                            const float* __restrict__ dt_bias,
                            const float* __restrict__ A_log,
                            const float* __restrict__ Ds,
                            float* __restrict__ yout)
{
    int tid = threadIdx.x;
    int d   = blockIdx.x * 256 + tid;
    int b   = blockIdx.y;
    __shared__ float s[64];

    for (int k = 0; k < CK; ++k) {
        float dtw[CR];
#pragma unroll
        for (int r = 0; r < CR; ++r) dtw[r] = dtw_all[(size_t)(k * CDI + d) * CR + r];
        float Av[CN];
#pragma unroll
        for (int i = 0; i < CN; ++i) Av[i] = -__expf(A_log[(size_t)(k * CDI + d) * CN + i]);
        float bias = dt_bias[k * CDI + d];
        float Dd   = Ds[k * CDI + d];
        float hst[CN];
#pragma unroll
        for (int i = 0; i < CN; ++i) hst[i] = 0.f;

        const float* xd = xdbl + (size_t)(b * CK + k) * CL * 64;

        for (int l = 0; l < CL; ++l) {
            __syncthreads();
            if (tid < 64) s[tid] = xd[(size_t)l * 64 + tid];
            __syncthreads();

            int ld = (k & 2) ? (1023 - l) : l;
            int hh = (k & 1) ? (ld & 31) : (ld >> 5);
            int ww = (k & 1) ? (ld >> 5) : (ld & 31);
            size_t pos = ((size_t)((b * 32 + hh) * 32 + ww) << 10) + d;

            float x_t = xconv[pos];
            float acc = bias;
#pragma unroll
            for (int r = 0; r < CR; ++r) acc += dtw[r] * s[r];
            float dt = fmaxf(acc, 0.f) + __logf(1.f + __expf(-fabsf(acc)));
            float dx = dt * x_t;
            float y = 0.f;
#pragma unroll
            for (int i = 0; i < CN; ++i) {
                float a = __expf(dt * Av[i]);
                hst[i] = a * hst[i] + dx * s[32 + i];
                y += hst[i] * s[48 + i];
            }
            float v = y + Dd * x_t;
            if (k == 0) yout[pos] = v;
            else        yout[pos] += v;
        }
        __syncthreads();
    }
}

__global__ void ln_gate_kernel(const float* __restrict__ y,
                               const float* __restrict__ xz,
                               const float* __restrict__ w,
                               const float* __restrict__ bvec,
                               float* __restrict__ out)
{
    int row = blockIdx.x;
    int tid = threadIdx.x;
    __shared__ float red[256];
    const float* yr = y + (size_t)row * CDI;

    float s = 0.f;
    for (int d = tid; d < CDI; d += 256) s += yr[d];
    red[tid] = s; __syncthreads();
    for (int o = 128; o > 0; o >>= 1) { if (tid < o) red[tid] += red[tid + o]; __syncthreads(); }
    float mean = red[0] / (float)CDI;
    __syncthreads();

    float s2 = 0.f;
    for (int d = tid; d < CDI; d += 256) { float t = yr[d] - mean; s2 += t * t; }
    red[tid] = s2; __syncthreads();
    for (int o = 128; o > 0; o >>= 1) { if (tid < o) red[tid] += red[tid + o]; __syncthreads(); }
    float inv = rsqrtf(red[0] / (float)CDI + 1e-5f);

    for (int d = tid; d < CDI; d += 256) {
        float t = (yr[d] - mean) * inv * w[d] + bvec[d];
        float z = xz[(size_t)row * 2048 + 1024 + d];
        out[(size_t)row * CDI + d] = t * (z / (1.f + __expf(-z)));
    }
}

// ---------------------------------------------------------------------------
extern "C" void kernel_launch(void* const* d_in, const int* in_sizes, int n_in,
                              void* d_out, int out_size, void* d_ws, size_t ws_size,
                              hipStream_t stream)
{
    (void)in_sizes; (void)n_in; (void)out_size; (void)ws_size;

    const float* x        = (const float*)d_in[0];
    const float* sigma    = (const float*)d_in[1];
    const float* tfw      = (const float*)d_in[2];
    const float* tinw     = (const float*)d_in[3];
    const float* mapInN   = (const float*)d_in[4];
    const float* mapOutN  = (const float*)d_in[11];
    const float* patchInW = (const float*)d_in[12];
    const float* outNorm  = (const float*)d_in[37];
    const float* patchOutW= (const float*)d_in[38];

    float* ws = (float*)d_ws;
    size_t off = 0;
    auto alloc = [&](size_t n) { float* p = ws + off; off += n; return p; };

    float* te    = alloc(CB * CMW);
    float* te2   = alloc(CB * CMW);
    float* mapA  = alloc(CB * CMW);
    float* mapB  = alloc(CB * CMW);
    float* mapU  = alloc(CB * 2 * CMFF);
    float* mapH  = alloc(CB * CMFF);
    float* cond  = alloc(CB * CMW);
    float* scale = alloc(CB * CDM);
    float* Am    = alloc((size_t)CROWS * 48);
    float* hfeat = alloc((size_t)CROWS * CDM);
    float* hn    = alloc((size_t)CROWS * CDM);
    float* xz    = alloc((size_t)CROWS * 2 * CDI);
    float* xconv = alloc((size_t)CROWS * CDI);
    float* xdbl  = alloc((size_t)CB * CK * CL * 64);
    float* yout  = alloc((size_t)CROWS * CDI);

    dim3 b256(256);

    // ---- time embedding + mapping net (tiny) ----
    time_embed_kernel<<<dim3(4), b256, 0, stream>>>(sigma, tfw, te);
    small_gemm_kernel<<<dim3((CB*CMW + 255)/256), b256, 0, stream>>>(
        te, tinw, nullptr, te2, CB, CMW, CMW, 0.f);
    rmsnorm_kernel<<<dim3(CB), b256, 0, stream>>>(te2, mapInN, mapA, CMW, 0);
    for (int mb = 0; mb < 2; ++mb) {
        const float* nrm = (const float*)d_in[5 + mb*3];
        const float* upw = (const float*)d_in[6 + mb*3];
        const float* dnw = (const float*)d_in[7 + mb*3];
        rmsnorm_kernel<<<dim3(CB), b256, 0, stream>>>(mapA, nrm, mapB, CMW, 0);
        small_gemm_kernel<<<dim3((CB*2*CMFF + 255)/256), b256, 0, stream>>>(
            mapB, upw, nullptr, mapU, CB, 2*CMFF, CMW, 0.f);
        glu_gelu_kernel<<<dim3((CB*CMFF + 255)/256), b256, 0, stream>>>(
            mapU, mapH, CB, CMFF);
        small_gemm_kernel<<<dim3((CB*CMW + 255)/256), b256, 0, stream>>>(
            mapH, dnw, mapA, mapA, CB, CMW, CMFF, 0.f);
    }
    rmsnorm_kernel<<<dim3(CB), b256, 0, stream>>>(mapA, mapOutN, cond, CMW, 0);

    // ---- patch merge -> hfeat (WMMA, K=48) ----
    patch_gather_kernel<<<dim3((CROWS*48 + 255)/256), b256, 0, stream>>>(x, Am);
    wmma_gemm_kernel<<<dim3(CDM/16, CROWS/16), dim3(32), 0, stream>>>(
        Am, patchInW, hfeat, CROWS, CDM, 48, 0);

    // ---- SS2D blocks ----
    for (int blk = 0; blk < 2; ++blk) {
        int base = 13 + blk * 12;
        const float* adaW    = (const float*)d_in[base + 0];
        const float* inProjW = (const float*)d_in[base + 1];
        const float* convW   = (const float*)d_in[base + 2];
        const float* convB   = (const float*)d_in[base + 3];
        const float* xProjW  = (const float*)d_in[base + 4];
        const float* dtProjW = (const float*)d_in[base + 5];
        const float* dtBias  = (const float*)d_in[base + 6];
        const float* Alog    = (const float*)d_in[base + 7];
        const float* DsP     = (const float*)d_in[base + 8];
        const float* onW     = (const float*)d_in[base + 9];
        const float* onB     = (const float*)d_in[base + 10];
        const float* outProjW= (const float*)d_in[base + 11];

        small_gemm_kernel<<<dim3((CB*CDM + 255)/256), b256, 0, stream>>>(
            cond, adaW, nullptr, scale, CB, CDM, CMW, 1.f);
        rmsnorm_kernel<<<dim3(CROWS), b256, 0, stream>>>(hfeat, scale, hn, CDM, CL);
        // xz = hn @ in_proj_w^T (512 -> 2048), TDM-staged WMMA
        wmma_gemm_tdm_kernel<<<dim3(2*CDI/NB, CROWS/MB), b256, 0, stream>>>(
            hn, inProjW, xz, CROWS, 2*CDI, CDM, 0);
        conv_silu_kernel<<<dim3((CB*CL*CDI + 255)/256), b256, 0, stream>>>(
            xz, convW, convB, xconv);
        xdbl_wmma_kernel<<<dim3(64/16, CROWS/16, CK), dim3(32), 0, stream>>>(
            xconv, xProjW, xdbl);
        scan_kernel<<<dim3(CDI/256, CB), b256, 0, stream>>>(
            xconv, xdbl, dtProjW, dtBias, Alog, DsP, yout);
        ln_gate_kernel<<<dim3(CROWS), b256, 0, stream>>>(yout, xz, onW, onB, xconv);
        // hfeat += gate @ out_proj_w^T (1024 -> 512), TDM-staged WMMA, acc
        wmma_gemm_tdm_kernel<<<dim3(CDM/NB, CROWS/MB), b256, 0, stream>>>(
            xconv, outProjW, hfeat, CROWS, CDM, CDI, 1);
    }

    // ---- final norm + patch split ----
    rmsnorm_kernel<<<dim3(CROWS), b256, 0, stream>>>(hfeat, outNorm, hn, CDM, 0);
    wmma_gemm_kernel<<<dim3(48/16, CROWS/16), dim3(32), 0, stream>>>(
        hn, patchOutW, Am, CROWS, 48, CDM, 0);
    patch_scatter_kernel<<<dim3((CROWS*48 + 255)/256), b256, 0, stream>>>(
        Am, (float*)d_out);
}